// _CandidateProjector_80771154968918
// MI455X (gfx1250) — compile-verified
//
#include <hip/hip_runtime.h>
#include <math.h>

#define NB 8
#define NH 256
#define NW 256
#define NHW (NH*NW)
#define NTOT (NB*NHW)
#define KTOP 80
#define CCAP 16384
#define PKSZ 400   // per-image peak block: [5][80] = val, -2xs, -2ys, xs^2+ys^2, inv2s

// ---- workspace layout (float offsets) ----
#define OFF_ROWOCC 0
#define OFF_ROWSC  (OFF_ROWOCC + NTOT)
#define OFF_PROD   (OFF_ROWSC  + NTOT)
#define OFF_SMAP   (OFF_PROD   + NTOT)
#define OFF_GAUSS  (OFF_SMAP   + NTOT)
#define OFF_CVAL   (OFF_GAUSS  + NTOT)
#define OFF_CIDX   (OFF_CVAL   + NB*CCAP)
#define OFF_PEAK   (OFF_CIDX   + NB*CCAP)
#define OFF_STATS  (OFF_PEAK + NB*PKSZ)
// stats per image (uint): [0]=prod_min [1]=prod_max [2]=g_min [3]=g_max [4]=cand_count

typedef float v2f  __attribute__((ext_vector_type(2)));
typedef float v8f  __attribute__((ext_vector_type(8)));
typedef unsigned int u32x4 __attribute__((ext_vector_type(4)));
typedef int   i32x4 __attribute__((ext_vector_type(4)));
typedef int   i32x8 __attribute__((ext_vector_type(8)));

__global__ void k_init(float* w) {
    unsigned int* st = (unsigned int*)(w + OFF_STATS);
    int t = threadIdx.x;
    if (t < NB) {
        st[t*8+0] = 0x7F7FFFFFu;  // prod min (FLT_MAX bits)
        st[t*8+1] = 0u;           // prod max (0.0 bits, values >= 0)
        st[t*8+2] = 0x7F7FFFFFu;  // gauss min
        st[t*8+3] = 0u;           // gauss max
        st[t*8+4] = 0u;           // candidate count
    }
}

// Horizontal 9-wide box sums of occ(score>0.2) and score (zero padding).
__global__ void k_rowbox(const float* __restrict__ score, float* w) {
    int row = blockIdx.x;            // b*NH + y
    int x = threadIdx.x;
    const float* img = score + row * NW;
    float so = 0.f, ss = 0.f;
    #pragma unroll
    for (int dx = -4; dx <= 4; ++dx) {
        int xx = x + dx;
        if ((unsigned)xx < NW) {
            float v = img[xx];
            ss += v;
            so += (v > 0.2f) ? 1.f : 0.f;
        }
    }
    w[OFF_ROWOCC + row*NW + x] = so;
    w[OFF_ROWSC  + row*NW + x] = ss;
}

// Vertical 9-wide box sums -> prod = avg9(occ)*avg9(score); per-image min/max.
__global__ void k_colbox(float* w) {
    int row = blockIdx.x;
    int b = row >> 8, y = row & 255;
    int x = threadIdx.x;
    float so = 0.f, ss = 0.f;
    #pragma unroll
    for (int dy = -4; dy <= 4; ++dy) {
        int yy = y + dy;
        if ((unsigned)yy < NH) {
            so += w[OFF_ROWOCC + (b*NH + yy)*NW + x];
            ss += w[OFF_ROWSC  + (b*NH + yy)*NW + x];
        }
    }
    float prod = (so * (1.0f/81.0f)) * (ss * (1.0f/81.0f));
    w[OFF_PROD + b*NHW + y*NW + x] = prod;

    float mn = prod, mx = prod;
    for (int o = 16; o > 0; o >>= 1) {
        mn = fminf(mn, __shfl_xor(mn, o, 32));
        mx = fmaxf(mx, __shfl_xor(mx, o, 32));
    }
    if ((threadIdx.x & 31) == 0) {
        unsigned int* st = (unsigned int*)(w + OFF_STATS);
        atomicMin(&st[b*8+0], __float_as_uint(mn));
        atomicMax(&st[b*8+1], __float_as_uint(mx));
    }
}

// score_map = score * norm01(prod)
__global__ void k_scoremap(const float* __restrict__ score, float* w) {
    int row = blockIdx.x;
    int b = row >> 8, y = row & 255;
    int x = threadIdx.x;
    const unsigned int* st = (const unsigned int*)(w + OFF_STATS);
    float mn = __uint_as_float(st[b*8+0]);
    float mx = __uint_as_float(st[b*8+1]);
    int idx = b*NHW + y*NW + x;
    float compact = (w[OFF_PROD + idx] - mn) / (mx - mn + 1e-6f);
    w[OFF_SMAP + idx] = score[idx] * compact;
}

// 5x5 max-pool NMS; positive local maxima appended to per-image candidate list.
__global__ void k_nms(float* w) {
    int row = blockIdx.x;
    int b = row >> 8, y = row & 255;
    int x = threadIdx.x;
    const float* sm = w + OFF_SMAP + b*NHW;
    float c = sm[y*NW + x];
    float mx = -1e30f;
    #pragma unroll
    for (int dy = -2; dy <= 2; ++dy) {
        int yy = y + dy;
        if ((unsigned)yy < NH) {
            #pragma unroll
            for (int dx = -2; dx <= 2; ++dx) {
                int xx = x + dx;
                if ((unsigned)xx < NW) mx = fmaxf(mx, sm[yy*NW + xx]);
            }
        }
    }
    if (c >= mx && c > 0.f) {
        unsigned int* st = (unsigned int*)(w + OFF_STATS);
        unsigned slot = atomicAdd(&st[b*8+4], 1u);
        if (slot < CCAP) {
            (w + OFF_CVAL)[b*CCAP + slot] = c;
            ((int*)(w + OFF_CIDX))[b*CCAP + slot] = y*NW + x;
        }
    }
}

// Per-image top-80 selection (val desc, idx asc) + derived per-peak terms.
__global__ void k_topk(const float* __restrict__ depth, float* w) {
    __shared__ float sval[256];
    __shared__ int   sidx[256];
    __shared__ int   sslot[256];
    int b = blockIdx.x;
    int t = threadIdx.x;
    unsigned int* st = (unsigned int*)(w + OFF_STATS);
    int cnt = (int)st[b*8+4];
    if (cnt > CCAP) cnt = CCAP;
    float* cv = w + OFF_CVAL + b*CCAP;
    int*   ci = ((int*)(w + OFF_CIDX)) + b*CCAP;
    float* pk = w + OFF_PEAK + b*PKSZ;   // [5][80] contiguous per image

    for (int k = 0; k < KTOP; ++k) {
        float bestv = 0.f; int besti = 0x7fffffff; int bestslot = -1;
        for (int i = t; i < cnt; i += 256) {
            float v = cv[i];
            if (v > 0.f) {
                int id = ci[i];
                if (v > bestv || (v == bestv && id < besti)) {
                    bestv = v; besti = id; bestslot = i;
                }
            }
        }
        sval[t] = bestv; sidx[t] = besti; sslot[t] = bestslot;
        for (int s = 128; s > 0; s >>= 1) {
            __syncthreads();
            if (t < s) {
                float v2 = sval[t+s]; int i2 = sidx[t+s];
                if (v2 > sval[t] || (v2 == sval[t] && i2 < sidx[t])) {
                    sval[t] = v2; sidx[t] = i2; sslot[t] = sslot[t+s];
                }
            }
        }
        __syncthreads();
        if (t == 0) {
            int slot = sslot[0];
            float val; int id;
            if (slot >= 0) { val = sval[0]; id = sidx[0]; cv[slot] = -1.f; }
            else           { val = 0.f;     id = 0; }
            float xs = (float)(id & (NW-1));
            float ys = (float)(id >> 8);
            float z = fmaxf(depth[b*NHW + id], 0.001f);
            float r = fminf(fmaxf(14.0f / z, 1.5f), 18.0f);
            float sg = 0.6f * r;
            float inv2 = 1.0f / (2.0f * sg * sg + 1e-6f);
            pk[0*KTOP + k] = val;
            pk[1*KTOP + k] = -2.0f * xs;
            pk[2*KTOP + k] = -2.0f * ys;
            pk[3*KTOP + k] = xs*xs + ys*ys;
            pk[4*KTOP + k] = inv2;
        }
        __syncthreads();
    }
}

// Gaussian splat: peak block staged into LDS via the Tensor Data Mover
// (tensor_load_to_lds + s_wait_tensorcnt), then per-tile d2 via
// V_WMMA_F32_16X16X4_F32 (M=16 peaks, N=16 pixels, K=4), VALU exp + max.
__global__ void __launch_bounds__(256) k_splat(float* w) {
    __shared__ float psh[PKSZ];        // [0:80)=val [80:160)=-2xs [160:240)=-2ys
                                       // [240:320)=xs^2+ys^2 [320:400)=inv2s
    __shared__ float aAx[5*32];        // pre-arranged per-lane WMMA A operand (K0/K2)
    __shared__ float aAy[5*32];        // pre-arranged per-lane WMMA A operand (K1/K3)
    int row = blockIdx.x;
    int b = row >> 8, y = row & 255;
    int t = threadIdx.x;
    const float* pkg = w + OFF_PEAK + b*PKSZ;

#if __has_builtin(__builtin_amdgcn_tensor_load_to_lds) && __has_builtin(__builtin_amdgcn_s_wait_tensorcnt)
    if (t < 32) {   // wave 0 issues one TDM 1-D copy: 400 floats -> LDS
        unsigned long long ga = (unsigned long long)(const void*)pkg;
        unsigned lds = (unsigned)(unsigned long long)(const void*)&psh[0];
        u32x4 g0;
        g0[0] = 1u;                                   // count=1 (valid descriptor)
        g0[1] = lds;                                  // lds_addr (bytes)
        g0[2] = (unsigned)(ga & 0xffffffffu);         // global_addr[31:0]
        g0[3] = ((unsigned)(ga >> 32) & 0x01ffffffu)  // global_addr[56:32]
              | 0x80000000u;                          // type=2 (image)
        i32x8 g1;
        g1[0] = 0x20000;                  // data_size=2 (4B), wg_mask=0, no flags
        g1[1] = (int)((PKSZ & 0xffff) << 16);         // tensor_dim0[15:0]
        g1[2] = (int)((PKSZ >> 16) | (1 << 16));      // tensor_dim0[31:16], tensor_dim1=1
        g1[3] = (int)(PKSZ << 16);                    // tensor_dim1 hi=0, tile_dim0=400
        g1[4] = 0;                                    // tile_dim1=0, tile_dim2=0
        g1[5] = PKSZ;                                 // tensor_dim0_stride[31:0]
        g1[6] = 0;                                    // stride hi, dim1_stride lo
        g1[7] = 0;
        i32x4 gz = {0, 0, 0, 0};
#if defined(__clang_major__) && __clang_major__ >= 23
        i32x8 gz8 = {0, 0, 0, 0, 0, 0, 0, 0};
        __builtin_amdgcn_tensor_load_to_lds(g0, g1, gz, gz, gz8, 0);
#else
        __builtin_amdgcn_tensor_load_to_lds(g0, g1, gz, gz, 0);
#endif
        __builtin_amdgcn_s_wait_tensorcnt(0);
    }
#else
    for (int i = t; i < PKSZ; i += 256) psh[i] = pkg[i];
#endif
    __syncthreads();

    // Pre-arrange A operand per (group g, lane): removes divergence in hot loop.
    // lane<16:  A[M=lane][K0]=-2xs, A[M=lane][K1]=-2ys
    // lane>=16: A[M=lane-16][K2]=1, A[M=lane-16][K3]=xs^2+ys^2
    if (t < 5*32) {
        int g = t >> 5, ln = t & 31, nn = ln & 15;
        aAx[t] = (ln < 16) ? psh[80  + g*16 + nn] : 1.0f;
        aAy[t] = (ln < 16) ? psh[160 + g*16 + nn] : psh[240 + g*16 + nn];
    }
    __syncthreads();

    int wv = t >> 5;
    int lane = t & 31;
    int n  = lane & 15;
    int hi = lane >> 4;
    float py = (float)y;

    float wavemn = 3.402823466e+38f, wavemx = 0.f;

    #pragma unroll
    for (int tile = 0; tile < 2; ++tile) {
        int x0 = wv*32 + tile*16;
        float px = (float)(x0 + n);
        // B matrix (4x16): rows [x, y, x^2+y^2, 1]
        v2f bv;
        bv.x = hi ? (px*px + py*py) : px;
        bv.y = hi ? 1.0f : py;
        float m = 0.f;
        #pragma unroll
        for (int g = 0; g < 5; ++g) {
            v2f av;
            av.x = aAx[g*32 + lane];
            av.y = aAy[g*32 + lane];
            v8f c = {};
            v8f d = __builtin_amdgcn_wmma_f32_16x16x4_f32(
                false, av, false, bv, (short)0, c, false, false);
            #pragma unroll
            for (int r = 0; r < 8; ++r) {
                int pm = g*16 + r + (hi << 3);
                float contrib = psh[pm] * __expf(-d[r] * psh[320 + pm]);
                m = fmaxf(m, contrib);
            }
        }
        m = fmaxf(m, __shfl_xor(m, 16, 32));
        if (hi == 0) w[OFF_GAUSS + b*NHW + y*NW + x0 + n] = m;
        wavemn = fminf(wavemn, m);
        wavemx = fmaxf(wavemx, m);
    }
    // reduce over 16 lanes (halves hold duplicates)
    for (int o = 8; o > 0; o >>= 1) {
        wavemn = fminf(wavemn, __shfl_xor(wavemn, o, 32));
        wavemx = fmaxf(wavemx, __shfl_xor(wavemx, o, 32));
    }
    if (lane == 0) {
        unsigned int* st = (unsigned int*)(w + OFF_STATS);
        atomicMin(&st[b*8+2], __float_as_uint(wavemn));
        atomicMax(&st[b*8+3], __float_as_uint(wavemx));
    }
}

__global__ void k_norm(const float* __restrict__ w, float* __restrict__ out) {
    int row = blockIdx.x;
    int b = row >> 8, y = row & 255;
    int x = threadIdx.x;
    const unsigned int* st = (const unsigned int*)(w + OFF_STATS);
    float mn = __uint_as_float(st[b*8+2]);
    float mx = __uint_as_float(st[b*8+3]);
    int idx = b*NHW + y*NW + x;
    out[idx] = (w[OFF_GAUSS + idx] - mn) / (mx - mn + 1e-6f);
}

extern "C" void kernel_launch(void* const* d_in, const int* in_sizes, int n_in,
                              void* d_out, int out_size, void* d_ws, size_t ws_size,
                              hipStream_t stream) {
    const float* score = (const float*)d_in[0];
    const float* depth = (const float*)d_in[1];
    float* out = (float*)d_out;
    float* w = (float*)d_ws;

    dim3 blk(256);
    dim3 grid(NB * NH);

    k_init    <<<1, 64, 0, stream>>>(w);
    k_rowbox  <<<grid, blk, 0, stream>>>(score, w);
    k_colbox  <<<grid, blk, 0, stream>>>(w);
    k_scoremap<<<grid, blk, 0, stream>>>(score, w);
    k_nms     <<<grid, blk, 0, stream>>>(w);
    k_topk    <<<NB, 256, 0, stream>>>(depth, w);
    k_splat   <<<grid, blk, 0, stream>>>(w);
    k_norm    <<<grid, blk, 0, stream>>>(w, out);
}